// AdaLoRAMLP_84121229459559
// MI455X (gfx1250) — compile-verified
//
#include <hip/hip_runtime.h>
#include <hip/hip_bf16.h>

// ---------------------------------------------------------------------------
// AdaLoRA MLP for MI455X (gfx1250, wave32, WMMA f32 16x16x4 + TDM staging)
//
// Sizes: B=16, T=1024, D=ADA=INTER=1024, RANK=8
//   ws layout (floats): xn[16*1024] | h[16*1024] | w[16*32768]
// ---------------------------------------------------------------------------

typedef float v2f __attribute__((ext_vector_type(2)));
typedef float v8f __attribute__((ext_vector_type(8)));
typedef unsigned int v4u __attribute__((ext_vector_type(4)));
typedef int v4i __attribute__((ext_vector_type(4)));
typedef int v8i __attribute__((ext_vector_type(8)));

#define FEAT 1024
#define RANK 8
#define NBATCH 16
#define NTOK 1024

__device__ __forceinline__ v8f wmma4(v2f a, v2f b, v8f c) {
  // D = A(16x4,f32) * B(4x16,f32) + C(16x16,f32)
  return __builtin_amdgcn_wmma_f32_16x16x4_f32(
      /*neg_a=*/false, a, /*neg_b=*/false, b,
      /*c_mod=*/(short)0, c, /*reuse_a=*/false, /*reuse_b=*/false);
}

__device__ __forceinline__ float gelu_exact(float v) {
  return 0.5f * v * (1.0f + erff(v * 0.70710678118654752f));
}

// ---------------------------------------------------------------------------
// TDM: contiguous copy of 8192 floats (32 KB) global -> LDS.
// D# built per CDNA5 ISA §8.3/8.4: 2D tile [8192,1], data_size=4B, type=2.
// Tracked with TENSORcnt; drain with s_wait_tensorcnt before use.
// This toolchain (clang-23/therock) uses the 6-arg builtin form:
//   (v4u g0, v8i g1, v4i g2, v4i g3, v8i extra, i32 cpol)
// ---------------------------------------------------------------------------
__device__ __forceinline__ void tdm_copy_8192f(const float* gsrc, float* ldsDst) {
  const unsigned long long ga = (unsigned long long)(uintptr_t)gsrc;
  const unsigned int la = (unsigned int)(uintptr_t)ldsDst;  // flat low32 = LDS addr
  const int E = 8192;

  v4u g0;
  g0[0] = 1u;                                   // count=1 (user descriptor)
  g0[1] = la;                                   // lds_addr
  g0[2] = (unsigned int)ga;                     // global_addr[31:0]
  g0[3] = ((unsigned int)(ga >> 32) & 0x01FFFFFFu) | (2u << 30);  // [56:32]|type=2

  v8i g1;
  g1[0] = (2 << 16);                 // workgroup_mask=0, data_size=2 (4 bytes)
  g1[1] = (E & 0xFFFF) << 16;        // tensor_dim0[15:0]  (bits 63:48)
  g1[2] = (E >> 16) | (1 << 16);     // tensor_dim0[31:16] | tensor_dim1[15:0]=1
  g1[3] = (E << 16);                 // tensor_dim1[31:16]=0 | tile_dim0=8192
  g1[4] = 1;                         // tile_dim1=1, tile_dim2=0 (unused)
  g1[5] = E;                         // tensor_dim0_stride[31:0]
  g1[6] = (E & 0xFFFF) << 16;        // stride0[47:32]=0 | tensor_dim1_stride[15:0]
  g1[7] = (E >> 16);                 // tensor_dim1_stride[47:16]

  v4i g2;
  g2[0] = 1;                         // tensor_dim2 = 1
  g2[1] = 1;                         // tensor_dim3 = 1
  g2[2] = E;                         // tensor_dim2_stride lo
  g2[3] = 0;                         // stride2 hi | tile_dim3=0

  v4i g3;
  g3[0] = E;                         // tensor_dim3_stride lo
  g3[1] = (1 << 16);                 // stride3 hi=0 | tensor_dim4[15:0]=1
  g3[2] = 0;                         // tensor_dim4 hi | tile_dim4=0
  g3[3] = 0;

  v8i gx = {0, 0, 0, 0, 0, 0, 0, 0}; // extra group (zero-filled, probe-verified)

  __builtin_amdgcn_tensor_load_to_lds(g0, g1, g2, g3, gx, /*cpol=*/0);
}

// ---------------------------------------------------------------------------
// Kernel 1: LayerNorm of ada_emb  [16,1024] -> xn
// ---------------------------------------------------------------------------
__global__ void ln_kernel(const float* __restrict__ ada,
                          const float* __restrict__ g,
                          const float* __restrict__ bta,
                          float* __restrict__ xn) {
  __shared__ float red[256];
  const int b = blockIdx.x;
  const int tid = threadIdx.x;
  const float* row = ada + b * FEAT;

  float s = 0.f;
  for (int i = tid; i < FEAT; i += 256) s += row[i];
  red[tid] = s; __syncthreads();
  for (int off = 128; off > 0; off >>= 1) {
    if (tid < off) red[tid] += red[tid + off];
    __syncthreads();
  }
  const float mu = red[0] * (1.0f / FEAT);
  __syncthreads();

  float v = 0.f;
  for (int i = tid; i < FEAT; i += 256) { float d = row[i] - mu; v += d * d; }
  red[tid] = v; __syncthreads();
  for (int off = 128; off > 0; off >>= 1) {
    if (tid < off) red[tid] += red[tid + off];
    __syncthreads();
  }
  const float rstd = rsqrtf(red[0] * (1.0f / FEAT) + 1e-5f);
  __syncthreads();

  for (int i = tid; i < FEAT; i += 256)
    xn[b * FEAT + i] = (row[i] - mu) * rstd * g[i] + bta[i];
}

// ---------------------------------------------------------------------------
// Kernel 2/3: C[16,N] = act(A[16,1024] @ B[1024,N] + bias)
// One wave -> one 16-column tile, K loop of 256 x V_WMMA_F32_16X16X4_F32.
// ---------------------------------------------------------------------------
template <bool GELU>
__global__ void hyper_gemm16(const float* __restrict__ A,
                             const float* __restrict__ B,
                             const float* __restrict__ bias,
                             float* __restrict__ C, int N) {
  const int wave = threadIdx.x >> 5;
  const int lane = threadIdx.x & 31;
  const int half = lane >> 4;
  const int n    = lane & 15;       // column within tile / A row
  const int n0   = (blockIdx.x * (blockDim.x >> 5) + wave) * 16;

  v8f acc = {};
  const float* arow = A + n * 1024;                 // A row m = lane&15
  const float* bcol = B + (size_t)2 * half * N + n0 + n;
  for (int k = 0; k < 1024; k += 4) {
    // prefetch the B rows 8 k-steps ahead (streams W2 through L2 once)
    __builtin_prefetch(bcol + (size_t)(k + 32) * N, 0, 1);
    v2f af = *(const v2f*)(arow + k + 2 * half);
    v2f bf;
    bf[0] = bcol[(size_t)(k + 0) * N];
    bf[1] = bcol[(size_t)(k + 1) * N];
    acc = wmma4(af, bf, acc);
  }
  const float bs = bias[n0 + n];
#pragma unroll
  for (int j = 0; j < 8; ++j) {
    const int m = j + 8 * half;     // C row
    float v = acc[j] + bs;
    if (GELU) v = gelu_exact(v);
    C[(size_t)m * N + n0 + n] = v;
  }
}

// ---------------------------------------------------------------------------
// Kernel 4: fused main path.
// Block = 128 threads (4 waves), handles 64 tokens of one batch.
// Factors staged in LDS (136 KB of 320 KB WGP):
//   bb1, bb2  [FEAT][RANK]  copied by the Tensor Data Mover (async, wave 0)
//   a1, a2    transposed to [RANK][FEAT] by VALU while the TDM runs
// Rank-8 B operands use duplicated column (n&7) instead of predicated zero:
// accumulator cols 8..15 hold duplicates and are simply never stored, so the
// inner loops are divergence-free (EXEC stays all-ones for WMMA).
// ---------------------------------------------------------------------------
__global__ __launch_bounds__(128) void adalora_main(
    const float* __restrict__ x, const float* __restrict__ w,
    float* __restrict__ out) {
  __shared__ __align__(16) float a1t[RANK * FEAT];   // transposed
  __shared__ __align__(16) float a2t[RANK * FEAT];   // transposed
  __shared__ __align__(16) float b1s[FEAT * RANK];
  __shared__ __align__(16) float b2s[FEAT * RANK];
  __shared__ __align__(16) float zstg[4][16 * 16];   // per-wave z tile
  __shared__ __align__(16) float y1s[4][16 * RANK];
  __shared__ __align__(16) float y2s[4][16 * RANK];

  const int b   = blockIdx.y;
  const int t0  = blockIdx.x * 64;
  const int tid = threadIdx.x;

  const float* wb = w + (size_t)b * (4 * FEAT * RANK);

  // Kick off async TDM copies of bb1/bb2 from wave 0.
  if (tid < 32) {
    tdm_copy_8192f(wb + FEAT * RANK, b1s);
    tdm_copy_8192f(wb + 3 * FEAT * RANK, b2s);
  }

  // Meanwhile all threads transpose-stage a1/a2.
  for (int i = tid * 4; i < FEAT * RANK; i += 128 * 4) {
    const int d = i >> 3, r = i & 7;   // i = d*8 + r, r in {0,4}
    float4 va = *(const float4*)&wb[i];
    float4 vc = *(const float4*)&wb[2 * FEAT * RANK + i];
    a1t[(r + 0) * FEAT + d] = va.x;
    a1t[(r + 1) * FEAT + d] = va.y;
    a1t[(r + 2) * FEAT + d] = va.z;
    a1t[(r + 3) * FEAT + d] = va.w;
    a2t[(r + 0) * FEAT + d] = vc.x;
    a2t[(r + 1) * FEAT + d] = vc.y;
    a2t[(r + 2) * FEAT + d] = vc.z;
    a2t[(r + 3) * FEAT + d] = vc.w;
  }

  // Wave 0 drains its TENSORcnt, then the block barrier publishes LDS.
  if (tid < 32) __builtin_amdgcn_s_wait_tensorcnt(0);
  __syncthreads();

  const int wv   = tid >> 5;
  const int lane = tid & 31;
  const int half = lane >> 4;
  const int n    = lane & 15;
  const int nd   = n & 7;            // duplicated rank column
  const int trow = t0 + wv * 16;
  const float* xrow = x + ((size_t)b * NTOK + trow) * FEAT;
  float* orow       = out + ((size_t)b * NTOK + trow) * FEAT;

  // ---- Stage 1: y1[16,8] = X[16,1024] @ a1[1024,8] -------------------------
  v8f c1 = {};
  {
    const float* xm  = xrow + (size_t)n * FEAT + 2 * half;  // A row m = lane&15
    const float* a1p = a1t + nd * FEAT + 2 * half;          // B col (dup n&7)
    for (int k = 0; k < 1024; k += 4) {
      v2f af = *(const v2f*)(xm + k);
      v2f bf = *(const v2f*)(a1p + k);
      c1 = wmma4(af, bf, c1);
    }
  }
  if (n < 8) {
#pragma unroll
    for (int j = 0; j < 8; ++j) y1s[wv][(j + 8 * half) * RANK + n] = c1[j];
  }
  __syncthreads();

  // ---- Stage 2+3: y2[16,8] = gelu(y1 @ bb1^T) @ a2 -------------------------
  v8f c2 = {};
  {
    const float* y1p = &y1s[wv][n * RANK + 2 * half];
    const float* a2p = a2t + nd * FEAT + 2 * half;
    for (int n0 = 0; n0 < FEAT; n0 += 16) {
      // z tile [16,16] = y1[16,8] @ bb1^T[8,16]
      const float* b1p = &b1s[(n0 + n) * RANK + 2 * half];
      v8f cz = {};
#pragma unroll
      for (int kk = 0; kk < 8; kk += 4) {
        v2f af = *(const v2f*)(y1p + kk);
        v2f bf = *(const v2f*)(b1p + kk);
        cz = wmma4(af, bf, cz);
      }
      // gelu, stage as A-layout source
#pragma unroll
      for (int j = 0; j < 8; ++j)
        zstg[wv][(j + 8 * half) * 16 + n] = gelu_exact(cz[j]);
      // y2 += z[16,16] @ a2[n0:n0+16, 0:8]  (dup col for n>=8)
      const float* zp = &zstg[wv][n * 16 + 2 * half];
#pragma unroll
      for (int kk = 0; kk < 16; kk += 4) {
        v2f af = *(const v2f*)(zp + kk);
        v2f bf = *(const v2f*)(a2p + n0 + kk);
        c2 = wmma4(af, bf, c2);
      }
    }
  }
  if (n < 8) {
#pragma unroll
    for (int j = 0; j < 8; ++j) y2s[wv][(j + 8 * half) * RANK + n] = c2[j];
  }
  __syncthreads();

  // ---- Stage 4: out = y2 @ bb2^T + x ---------------------------------------
  {
    const float* y2p = &y2s[wv][n * RANK + 2 * half];
    for (int n0 = 0; n0 < FEAT; n0 += 16) {
      v8f co;
#pragma unroll
      for (int j = 0; j < 8; ++j)
        co[j] = xrow[(size_t)(j + 8 * half) * FEAT + n0 + n];  // residual as C
      const float* b2p = &b2s[(n0 + n) * RANK + 2 * half];
#pragma unroll
      for (int kk = 0; kk < 8; kk += 4) {
        v2f af = *(const v2f*)(y2p + kk);
        v2f bf = *(const v2f*)(b2p + kk);
        co = wmma4(af, bf, co);
      }
#pragma unroll
      for (int j = 0; j < 8; ++j)
        orow[(size_t)(j + 8 * half) * FEAT + n0 + n] = co[j];
    }
  }
}

// ---------------------------------------------------------------------------
extern "C" void kernel_launch(void* const* d_in, const int* in_sizes, int n_in,
                              void* d_out, int out_size, void* d_ws, size_t ws_size,
                              hipStream_t stream) {
  const float* x    = (const float*)d_in[0];  // [16,1024,1024]
  const float* ada  = (const float*)d_in[1];  // [16,1024]
  const float* ln_g = (const float*)d_in[2];  // [1024]
  const float* ln_b = (const float*)d_in[3];  // [1024]
  const float* W1   = (const float*)d_in[4];  // [1024,1024]
  const float* b1   = (const float*)d_in[5];  // [1024]
  const float* W2   = (const float*)d_in[6];  // [1024,32768]
  const float* b2   = (const float*)d_in[7];  // [32768]
  float* out        = (float*)d_out;          // [16,1024,1024]

  float* ws = (float*)d_ws;
  float* xn = ws;                        // 16*1024
  float* h  = ws + NBATCH * FEAT;        // 16*1024
  float* wf = ws + 2 * NBATCH * FEAT;    // 16*32768

  // 1) LayerNorm
  ln_kernel<<<NBATCH, 256, 0, stream>>>(ada, ln_g, ln_b, xn);

  // 2) h = gelu(xn @ W1 + b1)   N=1024 : 8 blocks x 8 waves x 16 cols
  hyper_gemm16<true><<<1024 / 128, 256, 0, stream>>>(xn, W1, b1, h, 1024);

  // 3) w = h @ W2 + b2          N=32768 : 256 blocks
  hyper_gemm16<false><<<32768 / 128, 256, 0, stream>>>(h, W2, b2, wf, 32768);

  // 4) fused per-token low-rank chain + residual
  dim3 grid(NTOK / 64, NBATCH);
  adalora_main<<<grid, 128, 0, stream>>>(x, wf, out);
}